// GCNRecommendationModel_78039555768704
// MI455X (gfx1250) — compile-verified
//
#include <hip/hip_runtime.h>

// ---------------------------------------------------------------------------
// GCN: h1 = relu(bn(scatter(XW1)+b1)); h2 = relu(bn(scatter(h1W2)+b2));
//      out = h2 @ Wfc + bfc.   N=50000, E=600000, D=H=128, O=64, fp32.
// GEMMs: V_WMMA_F32_16X16X4_F32, W panel staged to LDS via TDM
// (tensor_load_to_lds + s_wait_tensorcnt). Scatter: native f32 atomics +
// global_prefetch of the next indirect row.
// ---------------------------------------------------------------------------

typedef __attribute__((ext_vector_type(2))) float v2f;
typedef __attribute__((ext_vector_type(8))) float v8f;
typedef __attribute__((ext_vector_type(4))) unsigned int v4u;
typedef __attribute__((ext_vector_type(8))) int v8i;
typedef __attribute__((ext_vector_type(4))) int v4i;

#define FDIM 128   // D == H == 128
#define BN_EPS 1e-5f

// ---------------- TDM: DMA a [rows x 16] fp32 panel (row stride = strideElems)
// from global into LDS (contiguous row-major), issued by the calling wave. ----
__device__ __forceinline__ void tdm_load_panel16(const float* gsrc, void* lds_dst,
                                                 int rows, int cols,
                                                 long strideElems) {
  unsigned long long ga = (unsigned long long)(uintptr_t)gsrc;
  unsigned int la = (unsigned int)(uintptr_t)lds_dst;   // low 32 bits = LDS offset

  v4u g0;
  g0[0] = 1u;                                            // count=1, user descriptor
  g0[1] = la;                                            // lds_addr
  g0[2] = (unsigned int)ga;                              // global_addr[31:0]
  g0[3] = (unsigned int)((ga >> 32) & 0x1FFFFFFu)        // global_addr[56:32]
        | (2u << 30);                                    // type=2 ("image")

  v8i g1;
  g1[0] = (int)(2u << 16);                               // data_size=4B; no mask/pad
  g1[1] = (int)((cols & 0xFFFF) << 16);                  // tensor_dim0[15:0]
  g1[2] = (int)(((cols >> 16) & 0xFFFF)                  // tensor_dim0[31:16]
        | ((rows & 0xFFFF) << 16));                      // tensor_dim1[15:0]
  g1[3] = (int)(((rows >> 16) & 0xFFFF)                  // tensor_dim1[31:16]
        | (16u << 16));                                  // tile_dim0 = 16
  g1[4] = rows;                                          // tile_dim1 = rows
  g1[5] = (int)(strideElems & 0xFFFFFFFFll);             // tensor_dim0_stride lo
  g1[6] = (int)((strideElems >> 32) & 0xFFFFll);         // stride hi | dim1_stride=0
  g1[7] = 0;

  v4i gz = {0, 0, 0, 0};                                 // 2-D tensor: groups 2/3 off
#if defined(__clang_major__) && (__clang_major__ >= 23)
  v8i gz8 = {0, 0, 0, 0, 0, 0, 0, 0};
  __builtin_amdgcn_tensor_load_to_lds(g0, g1, gz, gz, gz8, 0);
#else
  __builtin_amdgcn_tensor_load_to_lds(g0, g1, gz, gz, 0);
#endif
}

// ---------------- utility: fill ----------------
__global__ void k_fill(float* __restrict__ p, float v, size_t n) {
  size_t i = (size_t)blockIdx.x * blockDim.x + threadIdx.x;
  size_t st = (size_t)gridDim.x * blockDim.x;
  for (; i < n; i += st) p[i] = v;
}

// ---------------- degree (deg pre-initialized to 1.0 for self-loops) -------
__global__ void k_deg(const int* __restrict__ col, float* __restrict__ deg, int E) {
  int i = blockIdx.x * blockDim.x + threadIdx.x;
  int st = gridDim.x * blockDim.x;
  for (; i < E; i += st) unsafeAtomicAdd(&deg[col[i]], 1.0f);
}

__global__ void k_rsqrt_inplace(float* __restrict__ p, int n) {
  int i = blockIdx.x * blockDim.x + threadIdx.x;
  int st = gridDim.x * blockDim.x;
  for (; i < n; i += st) p[i] = rsqrtf(p[i]);   // deg >= 1 always (self-loop)
}

// ---------------- fp32 WMMA GEMM:  C[M x Ncols] = A[M x K] @ W[K x Ncols] --
// One 16x16 C tile per wave via V_WMMA_F32_16X16X4_F32 (K step 4).
// 8 waves per block share the K x 16 W panel, staged in LDS by TDM.
// blockIdx.x = column tile, blockIdx.y = group of 8 row tiles.
__global__ __launch_bounds__(256)
void k_gemm_wmma(const float* __restrict__ A, const float* __restrict__ W,
                 const float* __restrict__ bias, float* __restrict__ C,
                 int M, int K, int Ncols) {
  __shared__ float lds_w[FDIM * 16];          // up to 128x16 panel (8 KB)
  const int n0 = blockIdx.x * 16;

  // Wave 0 issues one TDM descriptor covering the whole K x 16 panel.
  if (threadIdx.x < 32) {
    tdm_load_panel16(W + n0, lds_w, K, Ncols, (long)Ncols);
    __builtin_amdgcn_s_wait_tensorcnt(0);
  }
  __syncthreads();

  const int wave = threadIdx.x >> 5;
  const int lane = threadIdx.x & 31;
  const int rt   = blockIdx.y * 8 + wave;     // row tile index (wave-uniform)
  if (rt * 16 >= M) return;                   // uniform exit: EXEC stays full

  const int mrow  = rt * 16 + (lane & 15);    // A row this lane feeds
  const int khalf = lane >> 4;                // 0: K={0,1}, 1: K={2,3} per step

  v8f acc = {};
  const float* arow = A + (size_t)mrow * K;
  for (int k = 0; k < K; k += 4) {
    v2f a, b;
    a[0] = arow[k + 2 * khalf + 0];
    a[1] = arow[k + 2 * khalf + 1];
    b[0] = lds_w[(k + 2 * khalf + 0) * 16 + (lane & 15)];
    b[1] = lds_w[(k + 2 * khalf + 1) * 16 + (lane & 15)];
    acc = __builtin_amdgcn_wmma_f32_16x16x4_f32(
        /*neg_a=*/false, a, /*neg_b=*/false, b,
        /*c_mod=*/(short)0, acc, /*reuse_a=*/false, /*reuse_b=*/false);
  }

  const int ncol = n0 + (lane & 15);
  const float bv = bias ? bias[ncol] : 0.0f;
  #pragma unroll
  for (int r = 0; r < 8; ++r) {               // lane holds C[m0+r+8*khalf][ncol]
    int m = rt * 16 + r + 8 * khalf;
    C[(size_t)m * Ncols + ncol] = acc[r] + bv;
  }
}

// ---------------- edge scatter: agg[col] += dinv[row]*dinv[col]*h[row] -----
// One wave per edge; each lane owns 4 consecutive channels (float4).
// Prefetch the next iteration's indirect h row to hide gather latency.
__global__ __launch_bounds__(256)
void k_scatter(const float* __restrict__ h, const int* __restrict__ row,
               const int* __restrict__ col, const float* __restrict__ dinv,
               float* __restrict__ agg, int E) {
  int wid = (blockIdx.x * blockDim.x + threadIdx.x) >> 5;
  int lane = threadIdx.x & 31;
  int nw = (gridDim.x * blockDim.x) >> 5;
  for (int e = wid; e < E; e += nw) {
    int en = e + nw;
    if (en < E) {                              // prefetch next indirect row
      int rn = row[en];
      __builtin_prefetch(h + (size_t)rn * FDIM + lane * 4, 0, 0);
    }
    int r = row[e], c = col[e];
    float coef = dinv[r] * dinv[c];
    float4 hv = ((const float4*)(h + (size_t)r * FDIM))[lane];
    float* dp = agg + (size_t)c * FDIM + lane * 4;
    unsafeAtomicAdd(dp + 0, coef * hv.x);
    unsafeAtomicAdd(dp + 1, coef * hv.y);
    unsafeAtomicAdd(dp + 2, coef * hv.z);
    unsafeAtomicAdd(dp + 3, coef * hv.w);
  }
}

// ---------------- fused self-loop + bias + BN statistics -------------------
// t[i,c] += dinv[i]^2 * h[i,c] + b[c]; stats[c]=sum, stats[128+c]=sumsq.
__global__ __launch_bounds__(256)
void k_post_conv(const float* __restrict__ h, float* __restrict__ t,
                 const float* __restrict__ b, const float* __restrict__ dinv,
                 float* __restrict__ stats, int N) {
  __shared__ float s_sum[256];
  __shared__ float s_sq[256];
  const int c = threadIdx.x & 127;
  const int half = threadIdx.x >> 7;          // two rows per block iteration
  const float bc = b[c];
  float lsum = 0.f, lsq = 0.f;
  for (int i = blockIdx.x * 2 + half; i < N; i += gridDim.x * 2) {
    float d = dinv[i];
    size_t idx = (size_t)i * FDIM + c;
    float v = t[idx] + d * d * h[idx] + bc;
    t[idx] = v;
    lsum += v;
    lsq  += v * v;
  }
  s_sum[threadIdx.x] = lsum;
  s_sq[threadIdx.x]  = lsq;
  __syncthreads();
  if (half == 0) {
    unsafeAtomicAdd(&stats[c],       s_sum[c] + s_sum[c + 128]);
    unsafeAtomicAdd(&stats[128 + c], s_sq[c]  + s_sq[c + 128]);
  }
}

// ---------------- BN apply + ReLU ------------------------------------------
__global__ __launch_bounds__(256)
void k_bn_relu(const float* __restrict__ t, const float* __restrict__ stats,
               const float* __restrict__ g, const float* __restrict__ beta,
               float* __restrict__ y, int N) {
  const int c = threadIdx.x & 127;
  const int half = threadIdx.x >> 7;
  const float inv_n = 1.0f / (float)N;
  const float mean  = stats[c] * inv_n;
  const float var   = stats[128 + c] * inv_n - mean * mean;
  const float sc    = g[c] * rsqrtf(var + BN_EPS);
  const float bb    = beta[c];
  for (int i = blockIdx.x * 2 + half; i < N; i += gridDim.x * 2) {
    size_t idx = (size_t)i * FDIM + c;
    float v = (t[idx] - mean) * sc + bb;
    y[idx] = v > 0.0f ? v : 0.0f;
  }
}

// ---------------------------------------------------------------------------
extern "C" void kernel_launch(void* const* d_in, const int* in_sizes, int n_in,
                              void* d_out, int out_size, void* d_ws, size_t ws_size,
                              hipStream_t stream) {
  (void)n_in; (void)out_size; (void)ws_size;

  const float* x    = (const float*)d_in[0];
  const int*   ei   = (const int*)  d_in[1];   // [2, E] row-major
  const float* W1   = (const float*)d_in[2];
  const float* b1   = (const float*)d_in[3];
  const float* g1   = (const float*)d_in[4];
  const float* bt1  = (const float*)d_in[5];
  const float* W2   = (const float*)d_in[6];
  const float* b2   = (const float*)d_in[7];
  const float* g2   = (const float*)d_in[8];
  const float* bt2  = (const float*)d_in[9];
  const float* Wfc  = (const float*)d_in[10];
  const float* bfc  = (const float*)d_in[11];
  float* out = (float*)d_out;

  const int N = in_sizes[0] / FDIM;            // 50000
  const int E = in_sizes[1] / 2;               // 600000
  const int O = 64;
  const int* erow = ei;                        // edge_index[0]
  const int* ecol = ei + E;                    // edge_index[1]

  // workspace layout (fp32): dinv | bufA (N*128) | bufB (N*128) | stats(256)
  float* ws = (float*)d_ws;
  const size_t nAl = ((size_t)N + 255) & ~(size_t)255;
  float* dinv  = ws;
  float* bufA  = dinv + nAl;
  float* bufB  = bufA + nAl * FDIM;
  float* stats = bufB + nAl * FDIM;

  const int T = 256;
  const size_t NH = (size_t)N * FDIM;
  const dim3 gemmGridH(FDIM / 16, (N / 16 + 7) / 8);   // 8 x 391
  const dim3 gemmGridO(O / 16,    (N / 16 + 7) / 8);   // 4 x 391

  // --- degree / normalization -------------------------------------------
  k_fill<<<1024, T, 0, stream>>>(dinv, 1.0f, (size_t)N);          // self-loop
  k_deg<<<2048, T, 0, stream>>>(ecol, dinv, E);
  k_rsqrt_inplace<<<1024, T, 0, stream>>>(dinv, N);

  // --- layer 1 -----------------------------------------------------------
  k_gemm_wmma<<<gemmGridH, T, 0, stream>>>(x, W1, nullptr, bufA, N, FDIM, FDIM);
  k_fill<<<2048, T, 0, stream>>>(bufB, 0.0f, NH);
  k_fill<<<1, T, 0, stream>>>(stats, 0.0f, 256);
  k_scatter<<<2048, T, 0, stream>>>(bufA, erow, ecol, dinv, bufB, E);
  k_post_conv<<<512, T, 0, stream>>>(bufA, bufB, b1, dinv, stats, N);
  k_bn_relu<<<512, T, 0, stream>>>(bufB, stats, g1, bt1, bufA, N);

  // --- layer 2 -----------------------------------------------------------
  k_gemm_wmma<<<gemmGridH, T, 0, stream>>>(bufA, W2, nullptr, bufB, N, FDIM, FDIM);
  k_fill<<<2048, T, 0, stream>>>(bufA, 0.0f, NH);
  k_fill<<<1, T, 0, stream>>>(stats, 0.0f, 256);
  k_scatter<<<2048, T, 0, stream>>>(bufB, erow, ecol, dinv, bufA, E);
  k_post_conv<<<512, T, 0, stream>>>(bufB, bufA, b2, dinv, stats, N);
  k_bn_relu<<<512, T, 0, stream>>>(bufA, stats, g2, bt2, bufB, N);

  // --- final FC (bias fused in GEMM epilogue) ---------------------------
  k_gemm_wmma<<<gemmGridO, T, 0, stream>>>(bufB, Wfc, bfc, out, N, FDIM, O);
}